// ClassicalSelfAttention_65481071405775
// MI455X (gfx1250) — compile-verified
//
#include <hip/hip_runtime.h>
#include <math.h>

typedef float v2f __attribute__((ext_vector_type(2)));
typedef float v8f __attribute__((ext_vector_type(8)));

#define BATCH 8
#define SEQ   2048
#define DIM   128
#define RSQRT_D 0.08838834764831845f   // 1/sqrt(128)
#define TK    32                        // t-rows staged per LDS chunk
#define PSTR  288                       // floats per row-pair in LDS (2*DIM + 32 pad)
                                        // PSTR % 64 == 32 -> hi/lo half-waves hit disjoint banks

// ---------------------------------------------------------------------------
// Kernel 1: per-row projections.  One wave32 per (b,t) row.
//   key[b,t] = x[b,t,:] . ent
//   a  [b,t] = (x[b,t,:] . (rot[:,0]+rot[:,1]+rot[:,2])) / sqrt(D)
// ---------------------------------------------------------------------------
__global__ __launch_bounds__(256)
void qsa_proj_kernel(const float* __restrict__ X, const float* __restrict__ rot,
                     const float* __restrict__ ent, float* __restrict__ keyv,
                     float* __restrict__ av) {
  const int row  = (blockIdx.x * blockDim.x + threadIdx.x) >> 5;  // global row id
  const int lane = threadIdx.x & 31;
  const int j    = lane * 4;                                       // 4 dims per lane

  float4 x = *(const float4*)(X + (size_t)row * DIM + j);

  float w0 = rot[3*j+0] + rot[3*j+1]  + rot[3*j+2];
  float w1 = rot[3*j+3] + rot[3*j+4]  + rot[3*j+5];
  float w2 = rot[3*j+6] + rot[3*j+7]  + rot[3*j+8];
  float w3 = rot[3*j+9] + rot[3*j+10] + rot[3*j+11];

  float dw = x.x*w0 + x.y*w1 + x.z*w2 + x.w*w3;
  float de = x.x*ent[j] + x.y*ent[j+1] + x.z*ent[j+2] + x.w*ent[j+3];

  #pragma unroll
  for (int off = 16; off >= 1; off >>= 1) {
    dw += __shfl_xor(dw, off, 32);
    de += __shfl_xor(de, off, 32);
  }
  if (lane == 0) {
    keyv[row] = de;
    av[row]   = dw * RSQRT_D;
  }
}

// ---------------------------------------------------------------------------
// Kernel 2: per-batch key min/max (gives the analytic softmax row-max:
//   max_t a*key_t = a>=0 ? a*kmax : a*kmin).
// ---------------------------------------------------------------------------
__global__ __launch_bounds__(256)
void qsa_minmax_kernel(const float* __restrict__ keyv,
                       float* __restrict__ kmax, float* __restrict__ kmin) {
  __shared__ float smax[256];
  __shared__ float smin[256];
  const int b = blockIdx.x, tid = threadIdx.x;
  float mx = -3.4e38f, mn = 3.4e38f;
  for (int t = tid; t < SEQ; t += 256) {
    float k = keyv[b * SEQ + t];
    mx = fmaxf(mx, k);
    mn = fminf(mn, k);
  }
  smax[tid] = mx; smin[tid] = mn;
  __syncthreads();
  for (int s = 128; s > 0; s >>= 1) {
    if (tid < s) {
      smax[tid] = fmaxf(smax[tid], smax[tid + s]);
      smin[tid] = fminf(smin[tid], smin[tid + s]);
    }
    __syncthreads();
  }
  if (tid == 0) { kmax[b] = smax[0]; kmin[b] = smin[0]; }
}

// ---------------------------------------------------------------------------
// Kernel 3: rank-1 attention, WMMA f32 16x16x4.
// Block = 128 threads (4 waves); block handles 64 s-rows of one batch,
// each wave one 16x128 output strip.  x[t:t+TK, :] is double-buffered in LDS
// with row pairs interleaved ((trow,col) -> pair*PSTR + col*2 + (trow&1)) so
// every B-fragment is one aligned 8-byte LDS load (fusible to 2addr_b64).
// Z (softmax denominator) is accumulated in the same loop.
// ---------------------------------------------------------------------------
__global__ __launch_bounds__(128)
void qsa_attn_kernel(const float* __restrict__ X, const float* __restrict__ keyv,
                     const float* __restrict__ av, const float* __restrict__ kmax,
                     const float* __restrict__ kmin, float* __restrict__ out) {
  __shared__ float xs[2][(TK / 2) * PSTR];    // 2 x 18 KB

  const int b     = blockIdx.x / (SEQ / 64);
  const int stile = blockIdx.x % (SEQ / 64);
  const int wave  = threadIdx.x >> 5;
  const int lane  = threadIdx.x & 31;
  const int s0    = stile * 64 + wave * 16;
  const int row   = lane & 15;
  const bool hi   = lane >= 16;
  const int koff  = hi ? 2 : 0;               // K-offset of this half-wave (A/B layout)

  const float* Xb   = X    + (size_t)b * SEQ * DIM;
  const float* keyb = keyv + b * SEQ;

  const float a = av[b * SEQ + s0 + row];
  const float m = (a >= 0.0f) ? a * kmax[b] : a * kmin[b];

  v8f acc[8] = {};
  float zacc = 0.0f;

  // chunk loader: 128 threads; thread -> row-pair p = tid>>3, 16-col segment
  const int lp = threadIdx.x >> 3;            // 0..15 : row pair within chunk
  const int lc = (threadIdx.x & 7) * 16;      // 0,16,...,112 : column base
  auto load_chunk = [&](int tc, int buf) {
    const float* s0p = Xb + (size_t)(tc + 2 * lp)     * DIM + lc;
    const float* s1p = Xb + (size_t)(tc + 2 * lp + 1) * DIM + lc;
    float* d = &xs[buf][lp * PSTR + lc * 2];
    #pragma unroll
    for (int q = 0; q < 4; ++q) {
      float4 ra = *(const float4*)(s0p + q * 4);
      float4 rb = *(const float4*)(s1p + q * 4);
      v2f* dq = (v2f*)(d + q * 8);
      v2f t0; t0.x = ra.x; t0.y = rb.x; dq[0] = t0;
      v2f t1; t1.x = ra.y; t1.y = rb.y; dq[1] = t1;
      v2f t2; t2.x = ra.z; t2.y = rb.z; dq[2] = t2;
      v2f t3; t3.x = ra.w; t3.y = rb.w; dq[3] = t3;
    }
  };

  load_chunk(0, 0);
  __syncthreads();

  int buf = 0;
  for (int tc = 0; tc < SEQ; tc += TK) {
    if (tc + TK < SEQ) load_chunk(tc + TK, buf ^ 1);

    const float* xl = xs[buf];
    #pragma unroll
    for (int tk = 0; tk < TK; tk += 4) {
      const int t = tc + tk;
      // probabilities this lane contributes (A-matrix layout: 2 K-slices/lane)
      const float p0 = __expf(fmaf(a, keyb[t + koff],     -m));
      const float p1 = __expf(fmaf(a, keyb[t + koff + 1], -m));
      zacc += p0 + p1;
      v2f afrag; afrag.x = p0; afrag.y = p1;

      // pair base for this half-wave: rows (tk+koff, tk+koff+1) interleaved
      const float* pl = xl + ((tk + koff) >> 1) * PSTR;
      #pragma unroll
      for (int n = 0; n < 8; ++n) {
        v2f bfrag = *(const v2f*)(pl + (n * 16 + row) * 2);  // one b64 LDS load
        acc[n] = __builtin_amdgcn_wmma_f32_16x16x4_f32(
            /*neg_a=*/false, afrag, /*neg_b=*/false, bfrag,
            /*c_mod=*/(short)0, acc[n], /*reuse_a=*/false, /*reuse_b=*/false);
      }
    }
    __syncthreads();   // next buffer filled AND everyone done with this one
    buf ^= 1;
  }

  // combine Z across the two half-waves: lane L now holds Z[L%16]
  zacc += __shfl_xor(zacc, 16, 32);

  float* ob = out + (size_t)b * SEQ * DIM;
  #pragma unroll
  for (int r = 0; r < 8; ++r) {
    const int M = r + (hi ? 8 : 0);                        // D-matrix row in acc[.][r]
    const float z   = __shfl(zacc, (hi ? 8 : 0) + r, 32);  // Z[M] via ds_bpermute
    const float inv = 1.0f / z;
    #pragma unroll
    for (int n = 0; n < 8; ++n) {
      ob[(size_t)(s0 + M) * DIM + n * 16 + row] = acc[n][r] * inv;
    }
  }
}

// ---------------------------------------------------------------------------
extern "C" void kernel_launch(void* const* d_in, const int* in_sizes, int n_in,
                              void* d_out, int out_size, void* d_ws, size_t ws_size,
                              hipStream_t stream) {
  const float* X   = (const float*)d_in[0];   // [8, 2048, 128] f32
  const float* rot = (const float*)d_in[1];   // [384] f32
  const float* ent = (const float*)d_in[2];   // [128] f32
  float*       out = (float*)d_out;           // [8, 2048, 128] f32

  float* keyv = (float*)d_ws;                 // B*S
  float* av   = keyv + BATCH * SEQ;           // B*S
  float* kmx  = av   + BATCH * SEQ;           // B
  float* kmn  = kmx  + BATCH;                 // B

  // 1) projections: one wave per row, 8 waves per block
  qsa_proj_kernel<<<dim3((BATCH * SEQ) / 8), dim3(256), 0, stream>>>(X, rot, ent, keyv, av);

  // 2) per-batch key min/max
  qsa_minmax_kernel<<<dim3(BATCH), dim3(256), 0, stream>>>(keyv, kmx, kmn);

  // 3) WMMA attention: 64 s-rows per block (4 waves x 16)
  qsa_attn_kernel<<<dim3(BATCH * (SEQ / 64)), dim3(128), 0, stream>>>(X, keyv, av, kmx, kmn, out);
}